// ChebGcnDecoder_28501402976715
// MI455X (gfx1250) — compile-verified
//
#include <hip/hip_runtime.h>

typedef float v2f __attribute__((ext_vector_type(2)));
typedef float v8f __attribute__((ext_vector_type(8)));

// ---------------------------------------------------------------------------
// V_WMMA_F32_16X16X4_F32: D(16x16 f32) = A(16x4 f32) x B(4x16 f32) + C
// (codegen-confirmed in rounds 1-2)
// ---------------------------------------------------------------------------
__device__ __forceinline__ v8f wmma_f32_16x16x4(v2f a, v2f b, v8f c) {
  return __builtin_amdgcn_wmma_f32_16x16x4_f32(
      false, a, false, b, (short)0, c, false, false);
}

// ---------------------------------------------------------------------------
// grid-stride zero fill
// ---------------------------------------------------------------------------
__global__ void k_zero(float* __restrict__ p, long long n) {
  long long i = (long long)blockIdx.x * blockDim.x + threadIdx.x;
  long long stride = (long long)gridDim.x * blockDim.x;
  for (; i < n; i += stride) p[i] = 0.0f;
}

// ---------------------------------------------------------------------------
// FC WMMA GEMM, specialized for M == 128:
//   block = 256 threads = 8 waves; wave w owns M-tile w.
//   block owns a 64-column N-stripe; W staged through LDS in KC=64 slabs,
//   shared by all 8 waves. W is stored K-pair-interleaved in LDS:
//     element (kk, c) -> lw[(kk>>1)*128 + 2*c + (kk&1)]
//   so each B-fragment (K pair at one column) is one contiguous 8-byte
//   ds_load_b64 into an even VGPR pair (no repack movs before the WMMA).
// Requires: M == 128, N % 64 == 0, K % 64 == 0. grid.x = N / 64.
// ---------------------------------------------------------------------------
__global__ __launch_bounds__(256) void k_gemm_wmma(
    const float* __restrict__ A, const float* __restrict__ W,
    const float* __restrict__ bias, float* __restrict__ OUT,
    int N, int K, int act) {
  __shared__ float lw[64 * 64];            // 16 KB slab
  const int tid  = threadIdx.x;
  const int wave = tid >> 5;
  const int lane = tid & 31;
  const int half = lane >> 4;
  const int lm   = lane & 15;
  const int n0   = blockIdx.x * 64;
  const int m0   = wave << 4;

  v8f acc[4];
#pragma unroll
  for (int nt = 0; nt < 4; ++nt) {
    float bv = bias[n0 + nt * 16 + lm];
#pragma unroll
    for (int r = 0; r < 8; ++r) acc[nt][r] = bv;
  }

  for (int kc = 0; kc < K; kc += 64) {
    __syncthreads();
    for (int i = tid; i < 64 * 64; i += 256) {
      int kk = i >> 6, c = i & 63;         // global-coalesced read of W row
      lw[((kk >> 1) << 7) + (c << 1) + (kk & 1)] =
          W[(long long)(kc + kk) * N + n0 + c];
    }
    __syncthreads();

    for (int kl = 0; kl < 64; kl += 4) {
      int ka = kl + (half << 1);           // even: kl or kl+2
      const float* arow = A + (long long)(m0 + lm) * K + kc + ka;
      v2f a; a.x = arow[0]; a.y = arow[1];
      const float* brow = lw + ((ka >> 1) << 7);
#pragma unroll
      for (int nt = 0; nt < 4; ++nt) {
        v2f b = *(const v2f*)(brow + ((nt * 16 + lm) << 1));
        acc[nt] = wmma_f32_16x16x4(a, b, acc[nt]);
      }
    }
  }

#pragma unroll
  for (int nt = 0; nt < 4; ++nt) {
#pragma unroll
    for (int r = 0; r < 8; ++r) {
      int row = m0 + r + (half << 3);
      float v = acc[nt][r];
      if (act) v = fmaxf(v, 0.0f);
      OUT[(long long)row * N + n0 + nt * 16 + lm] = v;
    }
  }
}

// ---------------------------------------------------------------------------
// Fused ChebConv WMMA GEMM:
//   OUT = store( lrelu( T0@W[0] + T1@W[1] + (2*P2 - T0)@W[2] + bias ) )
//   TN = compile-time N-tile count (accumulators stay in VGPRs).
//   All three W matrices staged through 48 KB LDS in KC=32 chunks using the
//   same K-pair-interleaved layout (single ds_load_b64 per B-fragment).
//   A-frags loaded once per K-step, reused across TN tiles: 3*TN WMMAs per
//   K-step per wave. Repeat(2,axis=1) + leaky-ReLU fused into the store.
// Requires: rows % 128 == 0. grid.x = rows / 128.
// ---------------------------------------------------------------------------
template <int TN>
__global__ __launch_bounds__(256) void k_cheb_gemm(
    const float* __restrict__ T0, const float* __restrict__ T1,
    const float* __restrict__ P2,
    const float* __restrict__ W, const float* __restrict__ bias,
    float* __restrict__ OUT,
    int K, int CO, int Nn, int doRepeat, int doLrelu) {
  __shared__ float lw[3 * 32 * 128];       // 48 KB max (KC=32, CO<=128)
  const int tid  = threadIdx.x;
  const int wave = tid >> 5;
  const int lane = tid & 31;
  const int half = lane >> 4;
  const int lm   = lane & 15;
  const int m0   = (blockIdx.x * 8 + wave) << 4;

  v8f acc[TN];
#pragma unroll
  for (int nt = 0; nt < TN; ++nt) {
    int col = nt * 16 + lm;
    float bv = (col < CO) ? bias[col] : 0.0f;
#pragma unroll
    for (int r = 0; r < 8; ++r) acc[nt][r] = bv;
  }

  const int KC   = (K >= 32) ? 32 : K;
  const int slab = KC * CO;                // floats per Chebyshev order
  const int CO2  = CO << 1;                // pair-row stride in LDS
  const long long WK = (long long)K * CO;

  for (int kc = 0; kc < K; kc += KC) {
    __syncthreads();
    for (int i = tid; i < 3 * slab; i += 256) {
      int t = i / slab;
      int rem = i - t * slab;
      int kk = rem / CO;
      int c  = rem - kk * CO;
      lw[t * slab + (kk >> 1) * CO2 + (c << 1) + (kk & 1)] =
          W[t * WK + (long long)(kc + kk) * CO + c];
    }
    __syncthreads();

    for (int kl = 0; kl < KC; kl += 4) {
      int ka = kl + (half << 1);           // even
      long long aoff = (long long)(m0 + lm) * K + kc + ka;
      v2f a0; a0.x = T0[aoff]; a0.y = T0[aoff + 1];
      v2f a1; a1.x = T1[aoff]; a1.y = T1[aoff + 1];
      v2f p2; p2.x = P2[aoff]; p2.y = P2[aoff + 1];
      v2f a2; a2.x = 2.0f * p2.x - a0.x; a2.y = 2.0f * p2.y - a0.y;

      const float* brow = lw + (ka >> 1) * CO2;
#pragma unroll
      for (int nt = 0; nt < TN; ++nt) {
        int col = nt * 16 + lm;
        bool colOK = (col < CO);
        int boff = (colOK ? col : 0) << 1;
        float m = colOK ? 1.0f : 0.0f;      // zero B-frag for padded cols
        v2f b0 = *(const v2f*)(brow + boff);
        v2f b1 = *(const v2f*)(brow + slab + boff);
        v2f b2 = *(const v2f*)(brow + 2 * slab + boff);
        b0.x *= m; b0.y *= m;
        b1.x *= m; b1.y *= m;
        b2.x *= m; b2.y *= m;
        acc[nt] = wmma_f32_16x16x4(a0, b0, acc[nt]);
        acc[nt] = wmma_f32_16x16x4(a1, b1, acc[nt]);
        acc[nt] = wmma_f32_16x16x4(a2, b2, acc[nt]);
      }
    }
  }

  // 16-row tile lies entirely inside one batch (Nn % 16 == 0)
  const int bb = m0 / Nn;
  const int nbase = m0 - bb * Nn;
#pragma unroll
  for (int nt = 0; nt < TN; ++nt) {
    int col = nt * 16 + lm;
    bool colOK = (col < CO);
#pragma unroll
    for (int r = 0; r < 8; ++r) {
      int rr = r + (half << 3);
      float v = acc[nt][r];
      if (doLrelu) v = (v >= 0.0f) ? v : 0.2f * v;
      if (colOK) {
        if (doRepeat) {
          long long o = ((long long)bb * (2 * Nn) + 2 * (nbase + rr)) * CO + col;
          OUT[o]      = v;
          OUT[o + CO] = v;
        } else {
          OUT[(long long)(m0 + rr) * CO + col] = v;
        }
      }
    }
  }
}

// ---------------------------------------------------------------------------
// deg[dst] += 1  (edge layout: (2, E) -> src = edge[e], dst = edge[E + e])
// ---------------------------------------------------------------------------
__global__ void k_degree(const int* __restrict__ edge, int E, float* __restrict__ deg) {
  int e = blockIdx.x * blockDim.x + threadIdx.x;
  if (e < E) atomicAdd(&deg[edge[E + e]], 1.0f);
}

// norm[e] = -dinv[src] * dinv[dst]
__global__ void k_norm(const int* __restrict__ edge, int E, const float* __restrict__ deg,
                       float* __restrict__ nrm) {
  int e = blockIdx.x * blockDim.x + threadIdx.x;
  if (e >= E) return;
  int s = edge[e], d = edge[E + e];
  float ds = deg[s], dd = deg[d];
  float is = (ds > 0.0f) ? rsqrtf(fmaxf(ds, 1.0f)) : 0.0f;
  float id = (dd > 0.0f) ? rsqrtf(fmaxf(dd, 1.0f)) : 0.0f;
  nrm[e] = -(is * id);
}

// ---------------------------------------------------------------------------
// Scatter propagate: dst_h[b, dst, c] += src_h[b, src, c] * norm[e]
// One block per edge; threads sweep B*C (coalesced over c).
// ---------------------------------------------------------------------------
__global__ void k_prop(const float* __restrict__ src_h, float* __restrict__ dst_h,
                       const int* __restrict__ edge, const float* __restrict__ nrm,
                       int E, int Nn, int C) {
  int e = blockIdx.x;
  int s = edge[e], d = edge[E + e];
  float w = nrm[e];
  int total = 128 * C;                      // B * C
  long long nodeStride = (long long)Nn * C;
  long long sbase = (long long)s * C;
  long long dbase = (long long)d * C;
  for (int i = threadIdx.x; i < total; i += blockDim.x) {
    int b = i / C;
    int c = i - b * C;
    atomicAdd(&dst_h[(long long)b * nodeStride + dbase + c],
              src_h[(long long)b * nodeStride + sbase + c] * w);
  }
}

// out[b, v, c] = H[b, perm[v], c], v < 6890, H is (128, 6912, 3)
__global__ void k_gather_out(const float* __restrict__ H, const int* __restrict__ perm,
                             float* __restrict__ OUT) {
  long long i = (long long)blockIdx.x * blockDim.x + threadIdx.x;
  long long total = 128LL * 6890 * 3;
  if (i >= total) return;
  int c = (int)(i % 3);
  long long t = i / 3;
  int v = (int)(t % 6890);
  int b = (int)(t / 6890);
  OUT[i] = H[((long long)b * 6912 + perm[v]) * 3 + c];
}

// ---------------------------------------------------------------------------
extern "C" void kernel_launch(void* const* d_in, const int* in_sizes, int n_in,
                              void* d_out, int out_size, void* d_ws, size_t ws_size,
                              hipStream_t stream) {
  (void)in_sizes; (void)n_in; (void)out_size; (void)ws_size;

  const float* x      = (const float*)d_in[0];
  const float* fc_w1  = (const float*)d_in[1];
  const float* fc_b1  = (const float*)d_in[2];
  const float* fc_w2  = (const float*)d_in[3];
  const float* fc_b2  = (const float*)d_in[4];
  const float* Wc[5]  = {(const float*)d_in[5],  (const float*)d_in[7],
                         (const float*)d_in[9],  (const float*)d_in[11],
                         (const float*)d_in[13]};
  const float* bcv[5] = {(const float*)d_in[6],  (const float*)d_in[8],
                         (const float*)d_in[10], (const float*)d_in[12],
                         (const float*)d_in[14]};
  const int* edges[4] = {(const int*)d_in[15], (const int*)d_in[16],
                         (const int*)d_in[17], (const int*)d_in[18]};
  const int* perm     = (const int*)d_in[19];
  float* out = (float*)d_out;

  const int GS[4]  = {864, 1728, 3456, 6912};
  const int CH[5]  = {128, 128, 64, 32, 16};
  const int COs[5] = {128, 64, 32, 16, 3};

  const long long BIG = 28311552LL;         // 128 * 6912 * 32 floats (113 MB)
  float* ws   = (float*)d_ws;
  float* bufA = ws;
  float* bufB = ws + BIG;                   // T1
  float* bufC = ws + 2 * BIG;               // P2 = prop(T1)
  float* bufD = ws + 3 * BIG;
  float* h1   = ws + 4 * BIG;               // 65536 floats
  float* deg  = h1 + 65536;                 // 6912 floats
  float* nrm  = deg + 6912;                 // 41472 floats

  // fc1: relu(x @ fc_w1 + fc_b1)  -> h1 (128, 512)
  k_gemm_wmma<<<512 / 64, 256, 0, stream>>>(x, fc_w1, fc_b1, h1, 512, 128, 1);
  // fc2: h1 @ fc_w2 + fc_b2 -> bufA == (128, 864, 128)
  k_gemm_wmma<<<110592 / 64, 256, 0, stream>>>(h1, fc_w2, fc_b2, bufA, 110592, 512, 0);

  float* curIn  = bufA;
  float* curOut = bufD;
  for (int layer = 0; layer < 5; ++layer) {
    int gi = (layer < 4) ? layer : 3;       // layer 4 reuses graph 3
    int Nn = GS[gi];
    int E  = 6 * Nn;
    int Ci = CH[layer];
    int CO = COs[layer];
    int doRepeat = (layer < 3) ? 1 : 0;
    int doLrelu  = (layer < 4) ? 1 : 0;
    const int* eidx = edges[gi];

    k_zero<<<32, 256, 0, stream>>>(deg, Nn);
    k_degree<<<(E + 255) / 256, 256, 0, stream>>>(eidx, E, deg);
    k_norm<<<(E + 255) / 256, 256, 0, stream>>>(eidx, E, deg, nrm);

    long long hsz = 128LL * Nn * Ci;
    k_zero<<<2048, 256, 0, stream>>>(bufB, hsz);
    k_prop<<<E, 256, 0, stream>>>(curIn, bufB, eidx, nrm, E, Nn, Ci);   // T1 = prop(T0)
    k_zero<<<2048, 256, 0, stream>>>(bufC, hsz);
    k_prop<<<E, 256, 0, stream>>>(bufB, bufC, eidx, nrm, E, Nn, Ci);    // P2 = prop(T1)

    int rows = 128 * Nn;
    int blocks = rows / 128;                // 8 M-tiles per block
    int tilesN = (CO + 15) / 16;
    switch (tilesN) {
      case 8:
        k_cheb_gemm<8><<<blocks, 256, 0, stream>>>(curIn, bufB, bufC, Wc[layer],
            bcv[layer], curOut, Ci, CO, Nn, doRepeat, doLrelu);
        break;
      case 4:
        k_cheb_gemm<4><<<blocks, 256, 0, stream>>>(curIn, bufB, bufC, Wc[layer],
            bcv[layer], curOut, Ci, CO, Nn, doRepeat, doLrelu);
        break;
      case 2:
        k_cheb_gemm<2><<<blocks, 256, 0, stream>>>(curIn, bufB, bufC, Wc[layer],
            bcv[layer], curOut, Ci, CO, Nn, doRepeat, doLrelu);
        break;
      default:
        k_cheb_gemm<1><<<blocks, 256, 0, stream>>>(curIn, bufB, bufC, Wc[layer],
            bcv[layer], curOut, Ci, CO, Nn, doRepeat, doLrelu);
        break;
    }

    float* t = curIn; curIn = curOut; curOut = t;
  }

  // curIn holds (128, 6912, 3); gather perm_back and slice to 6890 vertices
  {
    long long total = 128LL * 6890 * 3;
    k_gather_out<<<(unsigned)((total + 255) / 256), 256, 0, stream>>>(curIn, perm, out);
  }
}